// NegativeHardestContrastiveLoss_30734785970683
// MI455X (gfx1250) — compile-verified
//
#include <hip/hip_runtime.h>
#include <hip/hip_bf16.h>

typedef __attribute__((ext_vector_type(2))) float v2f;
typedef __attribute__((ext_vector_type(8))) float v8f;

#define C_DIM   64
#define HW_N    589824            // 768*768
#define ROWS    64                // NUM_NEG_PAIRS
#define TOPK    16
#define G_BLOCKS 144
#define COLS_PER_BLOCK 4096       // HW_N / G_BLOCKS
#define SLAB_W  128
#define NSLABS  32                // COLS_PER_BLOCK / SLAB_W
#define STRIDE  132               // 132 % 64 == 4 -> conflict-free tile stores
#define CAND_PER_ROW (G_BLOCKS * TOPK)  // 2304

__device__ __constant__ float kBig = 3.0e37f;

// ---------------- prep: gather sel (layout sel[c*64+p]) and snorm[p] ----------------
__global__ __launch_bounds__(256) void prep_kernel(const float* __restrict__ f1,
                                                   const int* __restrict__ neg,
                                                   float* __restrict__ selg,
                                                   float* __restrict__ snorm) {
  int tid = threadIdx.x;
  for (int j = tid; j < C_DIM * ROWS; j += 256) {
    int c = j >> 6, p = j & 63;
    selg[j] = f1[(size_t)c * HW_N + (size_t)(unsigned)neg[p]];
  }
  if (tid < ROWS) {
    size_t idx = (size_t)(unsigned)neg[tid];
    float s = 0.f;
    for (int c = 0; c < C_DIM; ++c) {
      float v = f1[(size_t)c * HW_N + idx];
      s += v * v;
    }
    snorm[tid] = s;
  }
}

// ---------------- phase A: WMMA distance tiles + per-block top-16 per row ----------------
__global__ __launch_bounds__(256) void phaseA_kernel(const float* __restrict__ f2,
                                                     const float* __restrict__ selg,
                                                     const float* __restrict__ snorm_g,
                                                     float* __restrict__ cand) {
  __shared__ float lds[12608];            // 50,432 bytes
  float* sel   = lds;                     // [4096]  packed A fragments (see below)
  float* snorm = lds + 4096;              // [64]
  float* slab  = lds + 4160;              // [64*132]

  const int tid  = threadIdx.x;
  const int lane = tid & 31;
  const int wv   = tid >> 5;              // 8 waves
  const int h    = lane >> 4;             // lane half
  const int m    = lane & 15;
  const int row  = tid >> 2;              // scan row: 0..63
  const int q    = tid & 3;               // scan quarter

  // Pack A fragments WMMA-ready: pair P = (kk*8 + t*2 + h)*16 + m holds
  //   { A[M=16t+m, c0], A[M=16t+m, c0+1] } with c0 = 4*kk + 2*h.
  // Float index of pair P is 2P = kk*256 + t*64 + h*32 + 2m -> one ds_load_b64
  // per WMMA, bank-conflict-free, no register repacking.
  for (int idx = tid; idx < 2048; idx += 256) {
    int m2  = idx & 15;
    int h2  = (idx >> 4) & 1;
    int t2  = (idx >> 5) & 3;
    int kk2 = idx >> 7;
    int c0  = 4 * kk2 + 2 * h2;
    int p   = 16 * t2 + m2;
    sel[idx * 2]     = selg[c0 * 64 + p];
    sel[idx * 2 + 1] = selg[(c0 + 1) * 64 + p];
  }
  if (tid < ROWS) snorm[tid] = snorm_g[tid];
  __syncthreads();

  // running per-thread top-16 (smallest) with tracked current max
  float top[TOPK];
  float cm = kBig;
  int   mi = 0;
#pragma unroll
  for (int j = 0; j < TOPK; ++j) top[j] = kBig;

  const size_t blockBase = (size_t)blockIdx.x * COLS_PER_BLOCK;
  const float* aLane = sel + h * 32 + 2 * m;   // per-lane base into packed A

  for (int s = 0; s < NSLABS; ++s) {
    // ---- compute: one 64x16 tile per wave via 64 chained WMMA f32 16x16x4 ----
    const size_t col0 = blockBase + (size_t)s * SLAB_W + (size_t)wv * 16;
    const float* f2p = f2 + col0 + m;     // per-lane column pointer
    v8f acc0 = {}, acc1 = {}, acc2 = {}, acc3 = {};
    float nrm = 0.f;
#pragma unroll 4
    for (int kk = 0; kk < 16; ++kk) {
      const int c0 = 4 * kk + 2 * h;      // K pair owned by this lane half
      float bx = f2p[(size_t)c0 * HW_N];
      float by = f2p[(size_t)(c0 + 1) * HW_N];
      v2f b; b.x = bx; b.y = by;
      nrm += bx * bx + by * by;
      const float* ak = aLane + kk * 256;
      v2f a0 = *(const v2f*)(ak);
      v2f a1 = *(const v2f*)(ak + 64);
      v2f a2 = *(const v2f*)(ak + 128);
      v2f a3 = *(const v2f*)(ak + 192);
      acc0 = __builtin_amdgcn_wmma_f32_16x16x4_f32(false, a0, false, b, (short)0, acc0, false, false);
      acc1 = __builtin_amdgcn_wmma_f32_16x16x4_f32(false, a1, false, b, (short)0, acc1, false, false);
      acc2 = __builtin_amdgcn_wmma_f32_16x16x4_f32(false, a2, false, b, (short)0, acc2, false, false);
      acc3 = __builtin_amdgcn_wmma_f32_16x16x4_f32(false, a3, false, b, (short)0, acc3, false, false);
    }
    // full column norm: combine both lane halves
    float fn = nrm + __shfl_xor(nrm, 16, 32);
    const int lc = wv * 16 + m;
#pragma unroll
    for (int v = 0; v < 8; ++v) {
      int p0 = 8 * h + v;
      float d0 = snorm[p0 +  0] + fn - 2.f * acc0[v];
      float d1 = snorm[p0 + 16] + fn - 2.f * acc1[v];
      float d2 = snorm[p0 + 32] + fn - 2.f * acc2[v];
      float d3 = snorm[p0 + 48] + fn - 2.f * acc3[v];
      slab[(p0 +  0) * STRIDE + lc] = fmaxf(d0, 0.f);
      slab[(p0 + 16) * STRIDE + lc] = fmaxf(d1, 0.f);
      slab[(p0 + 32) * STRIDE + lc] = fmaxf(d2, 0.f);
      slab[(p0 + 48) * STRIDE + lc] = fmaxf(d3, 0.f);
    }
    __syncthreads();

    // ---- prefetch next slab's f2 tile while the VALU-only scan runs ----
    if (s + 1 < NSLABS) {
      const float* pf = f2 + (size_t)row * HW_N + blockBase
                      + (size_t)(s + 1) * SLAB_W + q * 32;
      __builtin_prefetch(pf, 0, 1);       // global_prefetch_b8: 256 lines = whole slab
    }

    // ---- scan: 4 threads/row stream 32 columns each through top-16 ----
    const float* srow = slab + row * STRIDE + q * 32;
#pragma unroll 4
    for (int i = 0; i < 32; ++i) {
      float v = srow[i];
      if (v < cm) {
#pragma unroll
        for (int j = 0; j < TOPK; ++j) if (j == mi) top[j] = v;
        cm = top[0]; mi = 0;
#pragma unroll
        for (int j = 1; j < TOPK; ++j) if (top[j] > cm) { cm = top[j]; mi = j; }
      }
    }
    __syncthreads();
  }

  // ---- in-block merge: 4 lists per row -> 16 candidates per row ----
  float* cb = slab;                       // reuse slab LDS (256*16 floats)
#pragma unroll
  for (int j = 0; j < TOPK; ++j) cb[tid * TOPK + j] = top[j];
  __syncthreads();
  if (q == 0) {
    float mt[TOPK];
    float mcm = kBig; int mmi = 0;
#pragma unroll
    for (int j = 0; j < TOPK; ++j) mt[j] = kBig;
    for (int qq = 0; qq < 4; ++qq) {
      for (int j = 0; j < TOPK; ++j) {
        float v = cb[(row * 4 + qq) * TOPK + j];
        if (v < mcm) {
#pragma unroll
          for (int k = 0; k < TOPK; ++k) if (k == mmi) mt[k] = v;
          mcm = mt[0]; mmi = 0;
#pragma unroll
          for (int k = 1; k < TOPK; ++k) if (mt[k] > mcm) { mcm = mt[k]; mmi = k; }
        }
      }
    }
    float* dst = cand + ((size_t)row * G_BLOCKS + blockIdx.x) * TOPK;
    for (int j = 0; j < TOPK; ++j) dst[j] = mt[j];
  }
}

// ---------------- phase B: per-row global top-16 over all candidates ----------------
__global__ __launch_bounds__(256) void phaseB_kernel(const float* __restrict__ cand,
                                                     float* __restrict__ rowsum) {
  __shared__ float vals[CAND_PER_ROW];
  __shared__ float red[256];
  __shared__ int   redi[256];
  const int tid = threadIdx.x;
  const int r   = blockIdx.x;
  const float* src = cand + (size_t)r * CAND_PER_ROW;
  for (int i = tid; i < CAND_PER_ROW; i += 256) vals[i] = src[i];
  __syncthreads();

  float sum = 0.f;
  for (int it = 0; it < TOPK; ++it) {
    float mv = kBig; int mii = 0;
    for (int i = tid; i < CAND_PER_ROW; i += 256) {
      float v = vals[i];
      if (v < mv) { mv = v; mii = i; }
    }
    red[tid] = mv; redi[tid] = mii;
    __syncthreads();
    for (int off = 128; off > 0; off >>= 1) {
      if (tid < off && red[tid + off] < red[tid]) {
        red[tid] = red[tid + off];
        redi[tid] = redi[tid + off];
      }
      __syncthreads();
    }
    if (tid == 0) { sum += red[0]; vals[redi[0]] = kBig; }
    __syncthreads();
  }
  if (tid == 0) rowsum[r] = sum;
}

// ---------------- final: deterministic fixed-order mean over rows ----------------
__global__ __launch_bounds__(64) void final_kernel(const float* __restrict__ rowsum,
                                                   float* __restrict__ out) {
  __shared__ float s[64];
  const int tid = threadIdx.x;
  s[tid] = rowsum[tid];
  __syncthreads();
  for (int off = 32; off > 0; off >>= 1) {
    if (tid < off) s[tid] += s[tid + off];
    __syncthreads();
  }
  if (tid == 0) out[0] = -s[0] * (1.0f / (float)(TOPK * ROWS));
}

extern "C" void kernel_launch(void* const* d_in, const int* in_sizes, int n_in,
                              void* d_out, int out_size, void* d_ws, size_t ws_size,
                              hipStream_t stream) {
  const float* f1  = (const float*)d_in[0];   // feats1 (1,64,768,768)
  const float* f2  = (const float*)d_in[1];   // feats2
  // d_in[2] = positive_pairs (unused by reference)
  const int*   neg = (const int*)d_in[3];     // negative_indices (64)

  float* ws     = (float*)d_ws;
  float* selg   = ws;                          // 4096 floats
  float* snorm  = ws + 4096;                   // 64 floats
  float* cand   = ws + 4160;                   // 64*144*16 = 147456 floats
  float* rowsum = ws + 4160 + (size_t)ROWS * G_BLOCKS * TOPK;  // 64 floats

  prep_kernel  <<<1,        256, 0, stream>>>(f1, neg, selg, snorm);
  phaseA_kernel<<<G_BLOCKS, 256, 0, stream>>>(f2, selg, snorm, cand);
  phaseB_kernel<<<ROWS,     256, 0, stream>>>(cand, rowsum);
  final_kernel <<<1,        64,  0, stream>>>(rowsum, (float*)d_out);
}